// GraphConvolution_91036126806428
// MI455X (gfx1250) — compile-verified
//
#include <hip/hip_runtime.h>

typedef float v2f __attribute__((ext_vector_type(2)));
typedef float v8f __attribute__((ext_vector_type(8)));

#define NNODES 50000
#define NEDGES 1600000
#define DDIM   256
#define NBATCH 4
#define MTOTAL (NBATCH * NNODES)   // 200000 rows, divisible by 32

// ---------------------------------------------------------------------------
// Stage 1: support[MTOTAL x 256] = X[MTOTAL x 256] * W[256 x 256]  (fp32 WMMA)
// Wave computes a 32x64 tile: 8 accumulators (2 M-blocks x 4 N-blocks).
// The 4 B fragments per K-step are reused by both M-blocks -> 10 loads per
// 8 WMMAs instead of 9 per 4.
// Block = 128 threads = 4 waves on one 32-row slab (n-groups 0..3).
// ---------------------------------------------------------------------------
__global__ __launch_bounds__(128)
void gemm_wmma_f32(const float* __restrict__ X,
                   const float* __restrict__ W,
                   float* __restrict__ S) {
  const int tid   = threadIdx.x;
  const int lane  = tid & 31;
  const int wave  = tid >> 5;              // n-group 0..3
  const int mTile = blockIdx.x * 32;
  const int n0    = wave * 64;

  const int l15 = lane & 15;
  const int hi  = lane >> 4;               // 0: lanes 0-15, 1: lanes 16-31
  const float* xrow0 = X + (size_t)(mTile + l15) * DDIM;        // rows m..m+15
  const float* xrow1 = xrow0 + (size_t)16 * DDIM;               // rows m+16..m+31

  v8f c00 = {}, c01 = {}, c02 = {}, c03 = {};  // M-block 0
  v8f c10 = {}, c11 = {}, c12 = {}, c13 = {};  // M-block 1

  for (int k0 = 0; k0 < DDIM; k0 += 4) {
    const int ka = k0 + hi * 2;
    // A fragments: lane l15 = row, components = K=ka, ka+1 (8B loads)
    v2f a0 = *(const v2f*)(xrow0 + ka);
    v2f a1 = *(const v2f*)(xrow1 + ka);
    // B fragments: component v -> K=ka+v, col = n0 + t*16 + l15
    const float* wr0 = W + (size_t)ka * DDIM + n0 + l15;  // K = ka
    const float* wr1 = wr0 + DDIM;                        // K = ka+1
    v2f b0, b1, b2, b3;
    b0.x = wr0[0];   b0.y = wr1[0];
    b1.x = wr0[16];  b1.y = wr1[16];
    b2.x = wr0[32];  b2.y = wr1[32];
    b3.x = wr0[48];  b3.y = wr1[48];

    c00 = __builtin_amdgcn_wmma_f32_16x16x4_f32(false, a0, false, b0, (short)0, c00, false, false);
    c01 = __builtin_amdgcn_wmma_f32_16x16x4_f32(false, a0, false, b1, (short)0, c01, false, false);
    c02 = __builtin_amdgcn_wmma_f32_16x16x4_f32(false, a0, false, b2, (short)0, c02, false, false);
    c03 = __builtin_amdgcn_wmma_f32_16x16x4_f32(false, a0, false, b3, (short)0, c03, false, false);
    c10 = __builtin_amdgcn_wmma_f32_16x16x4_f32(false, a1, false, b0, (short)0, c10, false, false);
    c11 = __builtin_amdgcn_wmma_f32_16x16x4_f32(false, a1, false, b1, (short)0, c11, false, false);
    c12 = __builtin_amdgcn_wmma_f32_16x16x4_f32(false, a1, false, b2, (short)0, c12, false, false);
    c13 = __builtin_amdgcn_wmma_f32_16x16x4_f32(false, a1, false, b3, (short)0, c13, false, false);
  }

  // D layout: VGPR r -> rows base+r (lanes 0-15) and base+r+8 (lanes 16-31)
  const int rbase0 = mTile + hi * 8;
  const int rbase1 = rbase0 + 16;
#pragma unroll
  for (int r = 0; r < 8; ++r) {
    float* srow0 = S + (size_t)(rbase0 + r) * DDIM + l15 + n0;
    srow0[0]  = c00[r];
    srow0[16] = c01[r];
    srow0[32] = c02[r];
    srow0[48] = c03[r];
    float* srow1 = S + (size_t)(rbase1 + r) * DDIM + l15 + n0;
    srow1[0]  = c10[r];
    srow1[16] = c11[r];
    srow1[32] = c12[r];
    srow1[48] = c13[r];
  }
}

// ---------------------------------------------------------------------------
// Stage 2: out[b,n,:] = bias[:]   (vectorized float4 fill)
// ---------------------------------------------------------------------------
__global__ __launch_bounds__(256)
void init_bias(const float4* __restrict__ b4, float4* __restrict__ out4, int total4) {
  int i = blockIdx.x * blockDim.x + threadIdx.x;
  if (i < total4) out4[i] = b4[i & 63];   // 64 float4 per 256-wide row
}

// ---------------------------------------------------------------------------
// Stage 3: out[b, rows[e], :] += vals[e] * support[b, cols[e], :]
// 64 lanes per (edge, batch); each lane handles a float4 (4 fp32 atomics).
// grid = (E/4, BATCH), block = 256 threads = 4 edges.
// ---------------------------------------------------------------------------
__global__ __launch_bounds__(256)
void scatter_edges(const int* __restrict__ rows,
                   const int* __restrict__ cols,
                   const float* __restrict__ vals,
                   const float* __restrict__ S,
                   float* __restrict__ out) {
  const int b  = blockIdx.y;
  const int q  = threadIdx.x & 63;        // quad 0..63  (cols q*4 .. q*4+3)
  const int el = threadIdx.x >> 6;        // edge-in-block 0..3
  const long e = (long)blockIdx.x * 4 + el;
  if (e >= NEDGES) return;

  const int   r = rows[e];
  const int   c = cols[e];
  const float v = vals[e];

  const float4 s = ((const float4*)(S + ((size_t)b * NNODES + c) * DDIM))[q];
  float* o = out + ((size_t)b * NNODES + r) * DDIM + q * 4;
  atomicAdd(o + 0, v * s.x);
  atomicAdd(o + 1, v * s.y);
  atomicAdd(o + 2, v * s.z);
  atomicAdd(o + 3, v * s.w);
}

extern "C" void kernel_launch(void* const* d_in, const int* in_sizes, int n_in,
                              void* d_out, int out_size, void* d_ws, size_t ws_size,
                              hipStream_t stream) {
  const float* X    = (const float*)d_in[0];  // [4, 50000, 256]
  const int*   rows = (const int*)  d_in[1];  // [1.6M]
  const int*   cols = (const int*)  d_in[2];  // [1.6M]
  const float* vals = (const float*)d_in[3];  // [1.6M]
  const float* W    = (const float*)d_in[4];  // [256, 256]
  const float* bias = (const float*)d_in[5];  // [256]
  float* out = (float*)d_out;                 // [4, 50000, 256]
  float* S   = (float*)d_ws;                  // support scratch: 204.8 MB

  // Stage 1: dense projection via fp32 WMMA
  gemm_wmma_f32<<<MTOTAL / 32, 128, 0, stream>>>(X, W, S);

  // Stage 2: out = bias
  const int total4 = NBATCH * NNODES * (DDIM / 4);
  init_bias<<<(total4 + 255) / 256, 256, 0, stream>>>(
      (const float4*)bias, (float4*)out, total4);

  // Stage 3: COO scatter-add
  dim3 sg((NEDGES + 3) / 4, NBATCH);
  scatter_edges<<<sg, 256, 0, stream>>>(rows, cols, vals, S, out);
}